// Policy_2817498546564
// MI455X (gfx1250) — compile-verified
//
#include <hip/hip_runtime.h>
#include <math.h>

typedef float v2f __attribute__((ext_vector_type(2)));
typedef float v8f __attribute__((ext_vector_type(8)));

#define NQ     20
#define NL     4
#define DIMA   14
#define NSTATE (1 << NQ)
#define TILE   16384          /* 2^14 amplitudes per tile -> 128KB LDS (re+im) */
#define TPB    256            /* 8 waves of 32 */
#define NBLK   (NSTATE / TILE)

/* ---------------- workspace layout (in floats) ----------------
 * stateRe[2^20] | stateIm[2^20] | U1[4][20] 2x2 gates (8f each)
 * U2[20] 2x2 gates | M1hi[4][re256,im256] | M2a[4][re256,im256]
 * M2b[re256,im256] | zblk[NBLK][20]
 * Bit convention: linear index bit b  <->  qubit q = 19-b.
 */
constexpr int OFF_SRE = 0;
constexpr int OFF_SIM = NSTATE;
constexpr int OFF_U1  = 2 * NSTATE;
constexpr int OFF_U2  = OFF_U1 + NL * NQ * 8;
constexpr int OFF_M1  = OFF_U2 + NQ * 8;
constexpr int OFF_M2A = OFF_M1 + NL * 512;
constexpr int OFF_M2B = OFF_M2A + NL * 512;
constexpr int OFF_ZB  = OFF_M2B + 512;

__device__ __forceinline__ int insert0(int i, int m) {
  return ((i & ~(m - 1)) << 1) | (i & (m - 1));
}

/* apply a fused complex 2x2 gate on local bit `bit` of the LDS tile.
 * bit>=1: vectorized 2 adjacent amplitudes per thread (ds b64 traffic).
 * `bit` is a compile-time constant at all call sites, branch folds away. */
__device__ __forceinline__ void gate2(float (&sre)[TILE], float (&sim)[TILE],
                                      int bit, const float* __restrict__ g) {
  const int m = 1 << bit;
  const float g00r = g[0], g00i = g[1], g01r = g[2], g01i = g[3];
  const float g10r = g[4], g10i = g[5], g11r = g[6], g11i = g[7];
  if (bit >= 1) {
    for (int i2 = threadIdx.x; i2 < (TILE >> 2); i2 += TPB) {
      const int i0 = insert0(i2 << 1, m);      /* even: adjacent pair i0,i0+1 */
      const int i1 = i0 | m;
      const v2f xr = *(const v2f*)&sre[i0], xi = *(const v2f*)&sim[i0];
      const v2f yr = *(const v2f*)&sre[i1], yi = *(const v2f*)&sim[i1];
      *(v2f*)&sre[i0] = g00r * xr - g00i * xi + g01r * yr - g01i * yi;
      *(v2f*)&sim[i0] = g00r * xi + g00i * xr + g01r * yi + g01i * yr;
      *(v2f*)&sre[i1] = g10r * xr - g10i * xi + g11r * yr - g11i * yi;
      *(v2f*)&sim[i1] = g10r * xi + g10i * xr + g11r * yi + g11i * yr;
    }
  } else {
    for (int i = threadIdx.x; i < (TILE >> 1); i += TPB) {
      const int i0 = insert0(i, m), i1 = i0 | m;
      const float xr = sre[i0], xi = sim[i0], yr = sre[i1], yi = sim[i1];
      sre[i0] = g00r * xr - g00i * xi + g01r * yr - g01i * yi;
      sim[i0] = g00r * xi + g00i * xr + g01r * yi + g01i * yr;
      sre[i1] = g10r * xr - g10i * xi + g11r * yr - g11i * yi;
      sim[i1] = g10r * xi + g10i * xr + g11r * yi + g11i * yr;
    }
  }
  __syncthreads();
}

/* CNOT (conditional swap) on local bits of the LDS tile; vectorized when
 * the lower involved bit >= 1 (all call sites except cnotg(0,13)). */
__device__ __forceinline__ void cnotg(float (&sre)[TILE], float (&sim)[TILE],
                                      int cbit, int tbit) {
  const int mc = 1 << cbit, mt = 1 << tbit;
  const int blo = (cbit < tbit) ? cbit : tbit;
  const int bhi = (cbit < tbit) ? tbit : cbit;
  const int mlo = 1 << blo, mhi = 1 << bhi;
  if (blo >= 1) {
    for (int i2 = threadIdx.x; i2 < (TILE >> 3); i2 += TPB) {
      const int t  = insert0(insert0(i2 << 1, mlo), mhi);
      const int i0 = t | mc, i1 = i0 | mt;
      const v2f r = *(const v2f*)&sre[i0];
      *(v2f*)&sre[i0] = *(const v2f*)&sre[i1];
      *(v2f*)&sre[i1] = r;
      const v2f s = *(const v2f*)&sim[i0];
      *(v2f*)&sim[i0] = *(const v2f*)&sim[i1];
      *(v2f*)&sim[i1] = s;
    }
  } else {
    for (int i = threadIdx.x; i < (TILE >> 2); i += TPB) {
      const int t  = insert0(insert0(i, mlo), mhi);
      const int i0 = t | mc, i1 = i0 | mt;
      const float r = sre[i0]; sre[i0] = sre[i1]; sre[i1] = r;
      const float s = sim[i0]; sim[i0] = sim[i1]; sim[i1] = s;
    }
  }
  __syncthreads();
}

/* Fused 16x16 complex block-gate on a 4-bit group of the tile via
 * V_WMMA_F32_16X16X4_F32.  Tile viewed as S(16 x 1024) with
 * addr(r,c) = r*RS + c*CS (compile-time strides).  OUT = M @ S as
 * 4 real GEMMs:  Re = Mre*Sre + (-Mim)*Sim ,  Im = Mre*Sim + Mim*Sre
 * A frag: row=lane&15, K = kc*4 + 2*(lane>>4) + {0,1}
 * B frag: col=lane&15, K rows striped with half-offset 2
 * C/D   : col=lane&15, rows j and j+8 per VGPR j
 * RS==1: B-fragment K-rows and C/D row pairs are LDS-adjacent -> b64. */
template <int RS, int CS>
__device__ __forceinline__ void gate16(float (&sre)[TILE], float (&sim)[TILE],
                                       const float* __restrict__ Mre,
                                       const float* __restrict__ Mim) {
  const int lane = threadIdx.x & 31;
  const int wave = threadIdx.x >> 5;
  const int half = lane >> 4;
  const int row  = lane & 15;
  v2f Are[4], Aim[4], An[4];
#pragma unroll
  for (int kc = 0; kc < 4; ++kc) {
    const int k0 = kc * 4 + half * 2;
    v2f ar, ai;
    ar.x = Mre[row * 16 + k0]; ar.y = Mre[row * 16 + k0 + 1];
    ai.x = Mim[row * 16 + k0]; ai.y = Mim[row * 16 + k0 + 1];
    Are[kc] = ar; Aim[kc] = ai; An[kc] = -ai;   /* negate for the -Mim*Sim term */
  }
#pragma unroll
  for (int t = 0; t < 64 / (TPB / 32); ++t) {
    const int cc = wave + t * (TPB / 32);       /* wave-private 16-col chunk */
    const int c  = cc * 16 + row;
    v8f aR = {}; v8f aI = {};
#pragma unroll
    for (int kc = 0; kc < 4; ++kc) {
      const int k0 = kc * 4 + half * 2;
      v2f br, bi;
      if (RS == 1) {
        br = *(const v2f*)&sre[k0 + c * CS];
        bi = *(const v2f*)&sim[k0 + c * CS];
      } else {
        br.x = sre[(k0    ) * RS + c * CS];
        br.y = sre[(k0 + 1) * RS + c * CS];
        bi.x = sim[(k0    ) * RS + c * CS];
        bi.y = sim[(k0 + 1) * RS + c * CS];
      }
      aR = __builtin_amdgcn_wmma_f32_16x16x4_f32(false, Are[kc], false, br, (short)0, aR, false, false);
      aR = __builtin_amdgcn_wmma_f32_16x16x4_f32(false, An[kc],  false, bi, (short)0, aR, false, false);
      aI = __builtin_amdgcn_wmma_f32_16x16x4_f32(false, Are[kc], false, bi, (short)0, aI, false, false);
      aI = __builtin_amdgcn_wmma_f32_16x16x4_f32(false, Aim[kc], false, br, (short)0, aI, false, false);
    }
    if (RS == 1) {
#pragma unroll
      for (int j = 0; j < 8; j += 2) {
        const int rr = j + half * 8;
        v2f wr; wr.x = aR[j]; wr.y = aR[j + 1];
        v2f wi; wi.x = aI[j]; wi.y = aI[j + 1];
        *(v2f*)&sre[rr + c * CS] = wr;
        *(v2f*)&sim[rr + c * CS] = wi;
      }
    } else {
#pragma unroll
      for (int j = 0; j < 8; ++j) {
        const int rr = j + half * 8;
        sre[rr * RS + c * CS] = aR[j];
        sim[rr * RS + c * CS] = aI[j];
      }
    }
  }
  __syncthreads();
}

/* ---- prep: fuse rotations into 2x2 gates and build 16x16 block matrices ---- */
__global__ __launch_bounds__(256) void k_prep(const float* __restrict__ x,
                                              const float* __restrict__ theta,
                                              const float* __restrict__ weight,
                                              float* __restrict__ ws) {
  const int tid = threadIdx.x;
  float* U1 = ws + OFF_U1;
  float* U2 = ws + OFF_U2;
  if (tid < NL * NQ) {               /* U1 = RZ(g) RY(b) RZ(a) */
    const int k = tid / NQ, q = tid % NQ;
    const float a = theta[k * 3 * NQ + q * 3 + 0];
    const float b = theta[k * 3 * NQ + q * 3 + 1];
    const float g = theta[k * 3 * NQ + q * 3 + 2];
    const float cb = cosf(0.5f * b), sb = sinf(0.5f * b);
    const float apg = 0.5f * (a + g), amg = 0.5f * (a - g);
    float* u = U1 + tid * 8;
    u[0] =  cb * cosf(apg); u[1] = -cb * sinf(apg);
    u[2] = -sb * cosf(amg); u[3] = -sb * sinf(amg);
    u[4] =  sb * cosf(amg); u[5] = -sb * sinf(amg);
    u[6] =  cb * cosf(apg); u[7] =  cb * sinf(apg);
  }
  if (tid < NQ) {                    /* U2 = RZ(w2q+1*x) RY(w2q*x), layer-invariant */
    const int q = tid;
    const float t1 = weight[2 * q] * x[q];
    const float t2 = weight[2 * q + 1] * x[q];
    const float c1 = cosf(0.5f * t1), s1 = sinf(0.5f * t1);
    const float c2 = cosf(0.5f * t2), s2 = sinf(0.5f * t2);
    float* u = U2 + q * 8;
    u[0] =  c1 * c2; u[1] = -c1 * s2;
    u[2] = -s1 * c2; u[3] =  s1 * s2;
    u[4] =  s1 * c2; u[5] =  s1 * s2;
    u[6] =  c1 * c2; u[7] =  c1 * s2;
  }
  __syncthreads();
  const int rp = tid >> 4, r = tid & 15;  /* one matrix entry per thread */
  for (int k = 0; k < NL; ++k) {
    { /* M1hi: bits16..19 (qubits 3..0): S1 tensor then CNOT c32,c21,c10 */
      int u = rp;
      u ^= ((u >> 1) & 1) << 0;  /* apply c(1,0) first: u = pi^{-1}(rp) */
      u ^= ((u >> 2) & 1) << 1;
      u ^= ((u >> 3) & 1) << 2;
      float er = 1.f, ei = 0.f;
      for (int j = 0; j < 4; ++j) {
        const float* gt = U1 + (k * NQ + (3 - j)) * 8;
        const int e = (((u >> j) & 1) * 2 + ((r >> j) & 1)) * 2;
        const float gr = gt[e], gi = gt[e + 1];
        const float nr = er * gr - ei * gi; ei = er * gi + ei * gr; er = nr;
      }
      float* M = ws + OFF_M1 + k * 512;
      M[rp * 16 + r] = er; M[256 + rp * 16 + r] = ei;
    }
    { /* M2a: bits0..3 (qubits 19..16): plain S1 tensor product */
      float er = 1.f, ei = 0.f;
      for (int j = 0; j < 4; ++j) {
        const float* gt = U1 + (k * NQ + (19 - j)) * 8;
        const int e = (((rp >> j) & 1) * 2 + ((r >> j) & 1)) * 2;
        const float gr = gt[e], gi = gt[e + 1];
        const float nr = er * gr - ei * gi; ei = er * gi + ei * gr; er = nr;
      }
      float* M = ws + OFF_M2A + k * 512;
      M[rp * 16 + r] = er; M[256 + rp * 16 + r] = ei;
    }
  }
  { /* M2b = S3(bits1..3) * CNOT(1,0)*CNOT(2,1)*CNOT(3,2) (layer-invariant) */
    int v = r;
    v ^= ((v >> 3) & 1) << 2;  /* apply c(3,2) first: v = rho(r) */
    v ^= ((v >> 2) & 1) << 1;
    v ^= ((v >> 1) & 1) << 0;
    float er = 0.f, ei = 0.f;
    if ((rp & 1) == (v & 1)) {
      er = 1.f;
      for (int j = 1; j < 4; ++j) {
        const float* gt = U2 + (19 - j) * 8;
        const int e = (((rp >> j) & 1) * 2 + ((v >> j) & 1)) * 2;
        const float gr = gt[e], gi = gt[e + 1];
        const float nr = er * gr - ei * gi; ei = er * gi + ei * gr; er = nr;
      }
    }
    float* M = ws + OFF_M2B;
    M[rp * 16 + r] = er; M[256 + rp * 16 + r] = ei;
  }
}

__global__ void k_init(float* __restrict__ ws) {
  const unsigned i4 = blockIdx.x * blockDim.x + threadIdx.x;
  if (i4 < NSTATE / 4) {
    float4 z; z.x = 0.f; z.y = 0.f; z.z = 0.f; z.w = 0.f;
    float4 one = z; one.x = 1.0f;
    *(float4*)&ws[OFF_SRE + 4 * i4] = (i4 == 0) ? one : z;
    *(float4*)&ws[OFF_SIM + 4 * i4] = z;
  }
}

/* ---- pass 1: strided tile over bits {0..3} U {10..19}.  local bit = b-6 for b>=10. ---- */
__global__ __launch_bounds__(TPB) void k_pass1(float* __restrict__ ws, int layer) {
  __shared__ float sre[TILE];
  __shared__ float sim[TILE];
  float* gRe = ws + OFF_SRE;
  float* gIm = ws + OFF_SIM;
  const float* U1 = ws + OFF_U1 + layer * NQ * 8;
  const float* U2 = ws + OFF_U2;
  const float* M1 = ws + OFF_M1 + layer * 512;
  const int mid = blockIdx.x;
  /* B128 loads: 4-float groups stay inside each contiguous 16-element run */
  for (int i4 = threadIdx.x; i4 < TILE / 4; i4 += TPB) {
    const int i = i4 << 2;
    const int G = ((i >> 4) << 10) | (mid << 4) | (i & 15);
    *(float4*)&sre[i] = *(const float4*)&gRe[G];
    *(float4*)&sim[i] = *(const float4*)&gIm[G];
  }
  __syncthreads();
  if (layer > 0) {                      /* deferred tail of previous layer */
    cnotg(sre, sim, 0, 13);             /* CNOT(q19 -> q0) : bits (0 -> 19) */
    gate2(sre, sim, 0,  U2 + 19 * 8);   /* S3 on qubit 19 (bit 0)  */
    gate2(sre, sim, 13, U2 + 0 * 8);    /* S3 on qubit 0  (bit 19) */
  }
#pragma unroll
  for (int b = 10; b <= 15; ++b)        /* S1 on bits 10..15 */
    gate2(sre, sim, b - 6, U1 + (19 - b) * 8);
  gate16<1024, 1>(sre, sim, M1, M1 + 256);   /* S1(16..19)+CNOT(19,18)(18,17)(17,16) */
#pragma unroll
  for (int b = 16; b >= 11; --b)        /* chain CNOT(b,b-1) down to (11,10) */
    cnotg(sre, sim, b - 6, b - 7);
#pragma unroll
  for (int b = 11; b <= 18; ++b)        /* S3 on bits 11..18 */
    gate2(sre, sim, b - 6, U2 + (19 - b) * 8);
  for (int i4 = threadIdx.x; i4 < TILE / 4; i4 += TPB) {
    const int i = i4 << 2;
    const int G = ((i >> 4) << 10) | (mid << 4) | (i & 15);
    *(float4*)&gRe[G] = *(const float4*)&sre[i];
    *(float4*)&gIm[G] = *(const float4*)&sim[i];
  }
}

/* ---- pass 2: contiguous tile over bits 0..13 ---- */
__global__ __launch_bounds__(TPB) void k_pass2(float* __restrict__ ws, int layer) {
  __shared__ float sre[TILE];
  __shared__ float sim[TILE];
  float* gRe = ws + OFF_SRE;
  float* gIm = ws + OFF_SIM;
  const float* U1  = ws + OFF_U1 + layer * NQ * 8;
  const float* U2  = ws + OFF_U2;
  const float* M2a = ws + OFF_M2A + layer * 512;
  const float* M2b = ws + OFF_M2B;
  const int base = blockIdx.x * TILE;
  for (int i4 = threadIdx.x; i4 < TILE / 4; i4 += TPB) {
    const int i = i4 << 2;
    *(float4*)&sre[i] = *(const float4*)&gRe[base + i];
    *(float4*)&sim[i] = *(const float4*)&gIm[base + i];
  }
  __syncthreads();
#pragma unroll
  for (int b = 4; b <= 9; ++b)          /* S1 on bits 4..9 */
    gate2(sre, sim, b, U1 + (19 - b) * 8);
  gate16<1, 16>(sre, sim, M2a, M2a + 256);   /* S1 on bits 0..3 */
#pragma unroll
  for (int b = 10; b >= 4; --b)         /* chain CNOT(10,9) .. (4,3) */
    cnotg(sre, sim, b, b - 1);
  gate16<1, 16>(sre, sim, M2b, M2b + 256);   /* CNOT(3,2)(2,1)(1,0) + S3(1..3) */
#pragma unroll
  for (int b = 4; b <= 10; ++b)         /* S3 on bits 4..10 */
    gate2(sre, sim, b, U2 + (19 - b) * 8);
  for (int i4 = threadIdx.x; i4 < TILE / 4; i4 += TPB) {
    const int i = i4 << 2;
    *(float4*)&gRe[base + i] = *(const float4*)&sre[i];
    *(float4*)&gIm[base + i] = *(const float4*)&sim[i];
  }
}

/* ---- final: deferred tail of layer 3, then per-qubit <Z> partials ---- */
__global__ __launch_bounds__(TPB) void k_final(float* __restrict__ ws) {
  __shared__ float sre[TILE];
  __shared__ float sim[TILE];
  float* gRe = ws + OFF_SRE;
  float* gIm = ws + OFF_SIM;
  const float* U2 = ws + OFF_U2;
  const int mid = blockIdx.x;
  for (int i4 = threadIdx.x; i4 < TILE / 4; i4 += TPB) {
    const int i = i4 << 2;
    const int G = ((i >> 4) << 10) | (mid << 4) | (i & 15);
    *(float4*)&sre[i] = *(const float4*)&gRe[G];
    *(float4*)&sim[i] = *(const float4*)&gIm[G];
  }
  __syncthreads();
  cnotg(sre, sim, 0, 13);
  gate2(sre, sim, 0,  U2 + 19 * 8);
  gate2(sre, sim, 13, U2 + 0 * 8);
  float acc[NQ];
#pragma unroll
  for (int b = 0; b < NQ; ++b) acc[b] = 0.f;
  for (int i = threadIdx.x; i < TILE; i += TPB) {
    const int G = ((i >> 4) << 10) | (mid << 4) | (i & 15);
    const float p = sre[i] * sre[i] + sim[i] * sim[i];
#pragma unroll
    for (int b = 0; b < NQ; ++b) acc[b] += ((G >> b) & 1) ? -p : p;
  }
  __syncthreads();
  for (int b = 0; b < NQ; ++b) {        /* deterministic block reduction */
    sre[threadIdx.x] = acc[b];
    __syncthreads();
    for (int s = TPB / 2; s > 0; s >>= 1) {
      if (threadIdx.x < s) sre[threadIdx.x] += sre[threadIdx.x + s];
      __syncthreads();
    }
    if (threadIdx.x == 0) ws[OFF_ZB + mid * NQ + b] = sre[0];
    __syncthreads();
  }
}

/* ---- head: z -> logits -> softmax (tiny) ---- */
__global__ void k_head(const float* __restrict__ ws, const float* __restrict__ w,
                       const float* __restrict__ bias, float* __restrict__ out) {
  if (threadIdx.x == 0 && blockIdx.x == 0) {
    float z[NQ];
    for (int b = 0; b < NQ; ++b) {
      float s = 0.f;
      for (int blk = 0; blk < NBLK; ++blk) s += ws[OFF_ZB + blk * NQ + b];
      z[19 - b] = s;                    /* bit b <-> qubit 19-b */
    }
    float lg[DIMA]; float mx = -3.4e38f;
    for (int j = 0; j < DIMA; ++j) {
      float s = bias[j];
      for (int q = 0; q < NQ; ++q) s += w[j * NQ + q] * z[q];
      lg[j] = s; mx = fmaxf(mx, s);
    }
    float den = 0.f;
    for (int j = 0; j < DIMA; ++j) { lg[j] = expf(lg[j] - mx); den += lg[j]; }
    for (int j = 0; j < DIMA; ++j) out[j] = lg[j] / den;
  }
}

extern "C" void kernel_launch(void* const* d_in, const int* in_sizes, int n_in,
                              void* d_out, int out_size, void* d_ws, size_t ws_size,
                              hipStream_t stream) {
  const float* x      = (const float*)d_in[0];
  const float* theta  = (const float*)d_in[1];
  const float* weight = (const float*)d_in[2];
  const float* q1w    = (const float*)d_in[3];
  const float* q1b    = (const float*)d_in[4];
  float* ws  = (float*)d_ws;
  float* out = (float*)d_out;
  (void)in_sizes; (void)n_in; (void)out_size; (void)ws_size;

  k_prep<<<1, 256, 0, stream>>>(x, theta, weight, ws);
  k_init<<<NSTATE / 4 / 256, 256, 0, stream>>>(ws);
  for (int k = 0; k < NL; ++k) {
    k_pass1<<<NBLK, TPB, 0, stream>>>(ws, k);
    k_pass2<<<NBLK, TPB, 0, stream>>>(ws, k);
  }
  k_final<<<NBLK, TPB, 0, stream>>>(ws);
  k_head<<<1, 32, 0, stream>>>(ws, q1w, q1b, out);
}